// DNCMDSAE_68736656605195
// MI455X (gfx1250) — compile-verified
//
#include <hip/hip_runtime.h>
#include <hip/hip_bf16.h>
#include <math.h>

// ---------------- problem constants (match reference) ----------------
#define MS    64      // controller hidden
#define NH    4       // read heads
#define NCELL 32      // memory cells
#define NV    16      // vocab
#define WWD   64      // word size
#define NBAT  32      // batch
#define NTIME 256     // timesteps
#define IFP   800     // interface vector padded (799 -> 800)
#define BPB   16      // batches per block (== WMMA M tile)
#define EPSF  1e-6f

typedef float v2f __attribute__((ext_vector_type(2)));
typedef float v8f __attribute__((ext_vector_type(8)));

// FP32 WMMA: D(16x16) = A(16x4) * B(4x16) + C    -> v_wmma_f32_16x16x4_f32
__device__ __forceinline__ v8f wmma4(v2f a, v2f b, v8f c) {
  return __builtin_amdgcn_wmma_f32_16x16x4_f32(false, a, false, b, (short)0, c,
                                               false, false);
}
__device__ __forceinline__ float sigm(float x) { return 1.0f / (1.0f + __expf(-x)); }
__device__ __forceinline__ float softp(float x) { return (x > 20.f) ? x : log1pf(__expf(x)); }
__device__ __forceinline__ v2f ldv2(const float* p) { return *(const v2f*)p; }

// ---------------- weight transpose + K-pair pack prep kernel ----------------
// src[R][C] row-major (R = out-dim n, C = in-dim k).
// dst packed so one WMMA B-fragment (two consecutive k for one n) is one b64:
//   dst[(k>>1)*(2*Rp) + n*2 + (k&1)] = src[n*C + k]   (n < R, else 0; C even)
__global__ void dnc_tr_pack2(const float* __restrict__ src, float* __restrict__ dst,
                             int R, int C, int Rp) {
  int i = blockIdx.x * blockDim.x + threadIdx.x;
  int n = C * Rp;
  if (i >= n) return;
  int c = i / Rp, r = i % Rp;
  float v = (r < R) ? src[r * C + c] : 0.0f;
  dst[(c >> 1) * (2 * Rp) + r * 2 + (c & 1)] = v;
}

// ---------------- persistent DNC kernel: 1 block = 16 batches ----------------
__global__ __launch_bounds__(256) void dnc_main(
    const int* __restrict__ tokens, const float* __restrict__ emb,
    const float* __restrict__ b_lstm, const float* __restrict__ b_if,
    const float* __restrict__ b_out, const float* __restrict__ b_fc,
    const float* __restrict__ wihT,   // packed [160][2*256]
    const float* __restrict__ whhT,   // packed [32][2*256]
    const float* __restrict__ wifT,   // packed [32][2*800]
    const float* __restrict__ woutT,  // packed [160][2*64]
    const float* __restrict__ wfcT,   // packed [32][2*16]
    float* __restrict__ wsblk,        // per-block: M 32768 | link 16384 | iface 12800
    float* __restrict__ out)          // [32][16][256]
{
  __shared__ __align__(16) float sh_h[BPB * MS];          // 4 KB
  __shared__ __align__(16) float sh_c[BPB * MS];          // 4 KB
  __shared__ __align__(16) float sh_reads[BPB * NH * WWD];// 16 KB
  __shared__ __align__(16) float sh_scr[BPB * 384];       // 24 KB phase-aliased scratch
  __shared__ __align__(16) float sh_rw[BPB * NH * NCELL]; // 8 KB
  __shared__ float sh_usage[BPB * NCELL];   // 2 KB
  __shared__ float sh_prec[BPB * NCELL];    // 2 KB
  __shared__ float sh_ww[BPB * NCELL];      // 2 KB
  __shared__ float sh_modes[BPB * NH * 3];
  __shared__ float sh_sharp[BPB * NH * 2];
  __shared__ float sh_free[BPB * NH];
  __shared__ float sh_scal[48];             // [b*2]=ag,[b*2+1]=wg, [32+b]=sum(ww)
  __shared__ int   sh_tok[BPB];

  const int tid  = threadIdx.x;
  const int wave = tid >> 5;
  const int lane = tid & 31;
  const int l    = lane & 15;   // row (batch) / column within tile
  const int half = lane >> 4;   // K-half select for A/B fragments
  const int b0   = blockIdx.x * BPB;

  float* Mb    = wsblk + (size_t)blockIdx.x * 61952;
  float* linkb = Mb + BPB * NCELL * WWD;       // +32768
  float* ifb   = linkb + BPB * NCELL * NCELL;  // +49152

  // ---- zero-init carry state (ws is poisoned by harness) ----
  for (int i = tid; i < BPB * NCELL * WWD; i += 256) Mb[i] = 0.f;
  for (int i = tid; i < BPB * NCELL * NCELL; i += 256) linkb[i] = 0.f;
  for (int i = tid; i < BPB * MS; i += 256) { sh_h[i] = 0.f; sh_c[i] = 0.f; }
  for (int i = tid; i < BPB * NH * WWD; i += 256) sh_reads[i] = 0.f;
  for (int i = tid; i < BPB * NH * NCELL; i += 256) sh_rw[i] = 0.f;
  for (int i = tid; i < BPB * NCELL; i += 256) {
    sh_usage[i] = 0.f; sh_prec[i] = 0.f; sh_ww[i] = 0.f;
  }
  __threadfence_block();
  __syncthreads();

  for (int t = 0; t < NTIME; ++t) {
    if (tid < BPB) sh_tok[tid] = tokens[(b0 + tid) * NTIME + t];
    // keep the weight stream warm in L2/WGP$
    __builtin_prefetch(wihT + ((unsigned)(t * 4096 + tid * 16) & 65535), 0, 1);
    __syncthreads();

    // ===== Phase 1: LSTM gates g[16][256] = [x|reads]@wihT + h@whhT + b =====
    {
      float* g = sh_scr;
      const int nt0 = wave * 2;
      for (int nt = nt0; nt < nt0 + 2; ++nt) {
        const int ncol = nt * 16 + l;
        v8f acc = {0.f, 0.f, 0.f, 0.f, 0.f, 0.f, 0.f, 0.f};
        const float* xrow = emb + sh_tok[l] * MS;     // embedded token, row l
        for (int kb = 0; kb < 64; kb += 4) {          // K 0..63: x
          int k0 = kb + 2 * half;
          v2f a  = ldv2(xrow + k0);
          v2f bf = ldv2(wihT + (k0 >> 1) * 512 + 2 * ncol);
          acc = wmma4(a, bf, acc);
        }
        for (int kb = 0; kb < 256; kb += 4) {         // K 64..319: prev reads
          int k0 = kb + 2 * half;
          v2f a  = ldv2(sh_reads + l * 256 + k0);
          v2f bf = ldv2(wihT + (32 + (k0 >> 1)) * 512 + 2 * ncol);
          acc = wmma4(a, bf, acc);
        }
        for (int kb = 0; kb < 64; kb += 4) {          // recurrent h @ whhT
          int k0 = kb + 2 * half;
          v2f a  = ldv2(sh_h + l * 64 + k0);
          v2f bf = ldv2(whhT + (k0 >> 1) * 512 + 2 * ncol);
          acc = wmma4(a, bf, acc);
        }
        float bias = b_lstm[ncol];
#pragma unroll
        for (int v = 0; v < 8; ++v)
          g[(v + 8 * half) * 256 + ncol] = acc[v] + bias;
      }
    }
    __syncthreads();

    // ===== Phase 2: LSTM elementwise =====
    for (int i = tid; i < BPB * MS; i += 256) {
      int b = i >> 6, m = i & 63;
      float gi = sh_scr[b * 256 + m],       gf = sh_scr[b * 256 + 64 + m];
      float gg = sh_scr[b * 256 + 128 + m], go = sh_scr[b * 256 + 192 + m];
      float cc = sigm(gf) * sh_c[i] + sigm(gi) * tanhf(gg);
      sh_c[i] = cc;
      sh_h[i] = sigm(go) * tanhf(cc);
    }
    __syncthreads();

    // ===== Phase 3: interface = h @ wifT + b_if -> ifb (L2-resident scratch) =====
    for (int nt = wave; nt < 50; nt += 8) {
      const int ncol = nt * 16 + l;
      v8f acc = {0.f, 0.f, 0.f, 0.f, 0.f, 0.f, 0.f, 0.f};
      for (int kb = 0; kb < 64; kb += 4) {
        int k0 = kb + 2 * half;
        v2f a  = ldv2(sh_h + l * 64 + k0);
        v2f bf = ldv2(wifT + (k0 >> 1) * 1600 + 2 * ncol);
        acc = wmma4(a, bf, acc);
      }
      float bias = (ncol < 799) ? b_if[ncol] : 0.f;
#pragma unroll
      for (int v = 0; v < 8; ++v)
        ifb[(v + 8 * half) * IFP + ncol] = acc[v] + bias;
    }
    __threadfence_block();
    __syncthreads();

    // ===== Phase 4: small interface params -> LDS =====
    // offsets: rkeys 0 | rbeta 256 | wkey 260 | wbeta 324 | erase 325 | wvec 389
    //          free 453 | ag 457 | wg 458 | modes 459 | rmask 471 | wmask 727 | sharp 791
    for (int i = tid; i < BPB * NH; i += 256) {
      int b = i >> 2, h = i & 3;
      const float* ifr = ifb + b * IFP;
      sh_free[i] = sigm(ifr[453 + h]);
      sh_sharp[i * 2 + 0] = 1.f + softp(ifr[791 + h * 2 + 0]);
      sh_sharp[i * 2 + 1] = 1.f + softp(ifr[791 + h * 2 + 1]);
      float m0 = ifr[459 + h * 3], m1 = ifr[459 + h * 3 + 1], m2 = ifr[459 + h * 3 + 2];
      float mx = fmaxf(m0, fmaxf(m1, m2));
      float e0 = __expf(m0 - mx), e1 = __expf(m1 - mx), e2 = __expf(m2 - mx);
      float inv = 1.f / (e0 + e1 + e2);
      sh_modes[i * 3 + 0] = e0 * inv; sh_modes[i * 3 + 1] = e1 * inv; sh_modes[i * 3 + 2] = e2 * inv;
    }
    if (tid < BPB) {
      sh_scal[tid * 2 + 0] = sigm(ifb[tid * IFP + 457]);  // ag
      sh_scal[tid * 2 + 1] = sigm(ifb[tid * IFP + 458]);  // wg
    }
    __syncthreads();

    // ===== Phase 5: retention psi + usage update (uses OLD rw, OLD ww) =====
    float* psi = sh_scr;          // [0,512)
    float* cwb = sh_scr + 512;    // [512,1024)
    float* alc = sh_scr + 1024;   // [1024,1536)
    for (int i = tid; i < BPB * NCELL; i += 256) {
      int b = i >> 5, n = i & 31;
      float p = 1.f;
#pragma unroll
      for (int h = 0; h < NH; ++h)
        p *= (1.f - sh_free[b * 4 + h] * sh_rw[(b * 4 + h) * 32 + n]);
      psi[i] = p;
      float u = sh_usage[i], w = sh_ww[i];
      sh_usage[i] = (u + w - u * w) * p;
    }
    __syncthreads();

    // ===== Phase 6: allocation (serial sort per batch) + write content scores =====
    if (tid < BPB) {
      float u[NCELL]; int id[NCELL];
      for (int n = 0; n < NCELL; ++n) { u[n] = sh_usage[tid * 32 + n]; id[n] = n; }
      for (int a = 1; a < NCELL; ++a) {     // stable insertion sort, ascending
        float uv = u[a]; int iv = id[a]; int j = a - 1;
        while (j >= 0 && u[j] > uv) { u[j + 1] = u[j]; id[j + 1] = id[j]; --j; }
        u[j + 1] = uv; id[j + 1] = iv;
      }
      float cp = 1.f;
      for (int j = 0; j < NCELL; ++j) { alc[tid * 32 + id[j]] = (1.f - u[j]) * cp; cp *= u[j]; }
    }
    for (int i = tid; i < BPB * NCELL; i += 256) {   // masked cosine vs OLD M
      int b = i >> 5, n = i & 31;
      const float* ifr = ifb + b * IFP;
      const float* Mr = Mb + (b * 32 + n) * 64;
      float dot = 0.f, nm = 0.f, nk = 0.f;
      for (int w = 0; w < WWD; ++w) {
        float mask = sigm(ifr[727 + w]);
        float kk = ifr[260 + w] * mask;
        float mm = Mr[w] * mask;
        dot += mm * kk; nm += mm * mm; nk += kk * kk;
      }
      cwb[i] = dot / (sqrtf(nm) * sqrtf(nk) + EPSF);
    }
    __syncthreads();
    if (tid < BPB) {                                 // softmax(beta * score)
      float beta = 1.f + softp(ifb[tid * IFP + 324]);
      float mx = -1e30f;
      for (int n = 0; n < NCELL; ++n) { float v = beta * cwb[tid * 32 + n]; cwb[tid * 32 + n] = v; mx = fmaxf(mx, v); }
      float s = 0.f;
      for (int n = 0; n < NCELL; ++n) { float e = __expf(cwb[tid * 32 + n] - mx); cwb[tid * 32 + n] = e; s += e; }
      float inv = 1.f / s;
      for (int n = 0; n < NCELL; ++n) cwb[tid * 32 + n] *= inv;
    }
    __syncthreads();

    // ===== Phase 7: write weighting =====
    for (int i = tid; i < BPB * NCELL; i += 256) {
      int b = i >> 5;
      float ag = sh_scal[b * 2 + 0], wg = sh_scal[b * 2 + 1];
      sh_ww[i] = wg * (ag * alc[i] + (1.f - ag) * cwb[i]);
    }
    __syncthreads();

    // ===== Phase 8: M deallocate + erase + write; sum(ww) =====
    for (int i = tid; i < BPB * NCELL * WWD; i += 256) {
      int b = i >> 11, rem = i & 2047, n = rem >> 6, w = rem & 63;
      const float* ifr = ifb + b * IFP;
      float e  = sigm(ifr[325 + w]);
      float vv = ifr[389 + w];
      float wn = sh_ww[b * 32 + n];
      float m  = Mb[i] * psi[b * 32 + n];
      Mb[i] = m * (1.f - wn * e) + wn * vv;
    }
    if (tid < BPB) {
      float s = 0.f;
      for (int n = 0; n < NCELL; ++n) s += sh_ww[tid * 32 + n];
      sh_scal[32 + tid] = s;
    }
    __threadfence_block();
    __syncthreads();

    // ===== Phase 9: link matrix update (uses OLD prec), then prec update =====
    for (int i = tid; i < BPB * NCELL * NCELL; i += 256) {
      int b = i >> 10, rem = i & 1023, n = rem >> 5, m = rem & 31;
      float wn = sh_ww[b * 32 + n], wm = sh_ww[b * 32 + m];
      float lv = (1.f - wn - wm) * linkb[i] + wn * sh_prec[b * 32 + m];
      linkb[i] = (n == m) ? 0.f : lv;
    }
    __threadfence_block();
    __syncthreads();
    for (int i = tid; i < BPB * NCELL; i += 256) {
      int b = i >> 5;
      sh_prec[i] = (1.f - sh_scal[32 + b]) * sh_prec[i] + sh_ww[i];
    }
    __syncthreads();

    // ===== Phase 10: forward/backward distributions (OLD rw) + sharpen =====
    float* fwb = sh_scr;            // [0,2048)
    float* bwb = sh_scr + 2048;     // [2048,4096)
    float* crb = sh_scr + 4096;     // [4096,6144)
    for (int i = tid; i < BPB * NH * NCELL; i += 256) {
      int b = i >> 7, rem = i & 127, h = rem >> 5, n = rem & 31;
      const float* lrow = linkb + (b * 32 + n) * 32;  // link[b][n][m]
      const float* lcol = linkb + b * 1024 + n;       // link[b][m][n]
      const float* rr   = sh_rw + (b * 4 + h) * 32;
      float f = 0.f, bk = 0.f;
      for (int m = 0; m < NCELL; ++m) {
        float r = rr[m];
        f  += lrow[m] * r;
        bk += lcol[m * 32] * r;
      }
      fwb[i] = __powf(f + EPSF, sh_sharp[(b * 4 + h) * 2 + 0]);
      bwb[i] = __powf(bk + EPSF, sh_sharp[(b * 4 + h) * 2 + 1]);
    }
    __syncthreads();
    if (tid < BPB * NH) {           // normalize fw/bw rows
      float sf = 0.f, sb = 0.f;
      for (int n = 0; n < NCELL; ++n) { sf += fwb[tid * 32 + n]; sb += bwb[tid * 32 + n]; }
      float invf = 1.f / sf, invb = 1.f / sb;
      for (int n = 0; n < NCELL; ++n) { fwb[tid * 32 + n] *= invf; bwb[tid * 32 + n] *= invb; }
    }
    // ===== Phase 11: read content addressing on NEW M =====
    for (int i = tid; i < BPB * NH * NCELL; i += 256) {
      int b = i >> 7, rem = i & 127, h = rem >> 5, n = rem & 31;
      const float* ifr = ifb + b * IFP;
      const float* Mr = Mb + (b * 32 + n) * 64;
      float dot = 0.f, nm = 0.f, nk = 0.f;
      for (int w = 0; w < WWD; ++w) {
        float mask = sigm(ifr[471 + h * 64 + w]);
        float kk = ifr[h * 64 + w] * mask;
        float mm = Mr[w] * mask;
        dot += mm * kk; nm += mm * mm; nk += kk * kk;
      }
      crb[i] = dot / (sqrtf(nm) * sqrtf(nk) + EPSF);
    }
    __syncthreads();
    if (tid < BPB * NH) {           // softmax(rbeta * score)
      int b = tid >> 2, h = tid & 3;
      float beta = 1.f + softp(ifb[b * IFP + 256 + h]);
      float mx = -1e30f;
      for (int n = 0; n < NCELL; ++n) { float v = beta * crb[tid * 32 + n]; crb[tid * 32 + n] = v; mx = fmaxf(mx, v); }
      float s = 0.f;
      for (int n = 0; n < NCELL; ++n) { float e = __expf(crb[tid * 32 + n] - mx); crb[tid * 32 + n] = e; s += e; }
      float inv = 1.f / s;
      for (int n = 0; n < NCELL; ++n) crb[tid * 32 + n] *= inv;
    }
    __syncthreads();

    // ===== Phase 12: combine rw, compute reads = rw @ M =====
    for (int i = tid; i < BPB * NH * NCELL; i += 256) {
      int b = i >> 7, rem = i & 127, h = rem >> 5;
      const float* md = sh_modes + (b * 4 + h) * 3;
      sh_rw[i] = md[0] * bwb[i] + md[1] * crb[i] + md[2] * fwb[i];
    }
    __syncthreads();
    for (int i = tid; i < BPB * NH * WWD; i += 256) {
      int b = i >> 8, rem = i & 255, h = rem >> 6, w = rem & 63;
      const float* rr = sh_rw + (b * 4 + h) * 32;
      const float* Mr = Mb + b * 2048 + w;
      float s = 0.f;
      for (int n = 0; n < NCELL; ++n) s += rr[n] * Mr[n * 64];
      sh_reads[i] = s;
    }
    __syncthreads();

    // ===== Phase 13: y = [h|reads]@woutT + b_out ; logits = y@wfcT + b_fc =====
    float* yb = sh_scr;             // reuse scratch
    if (wave < 4) {
      const int ncol = wave * 16 + l;
      v8f acc = {0.f, 0.f, 0.f, 0.f, 0.f, 0.f, 0.f, 0.f};
      for (int kb = 0; kb < 64; kb += 4) {
        int k0 = kb + 2 * half;
        v2f a  = ldv2(sh_h + l * 64 + k0);
        v2f bf = ldv2(woutT + (k0 >> 1) * 128 + 2 * ncol);
        acc = wmma4(a, bf, acc);
      }
      for (int kb = 0; kb < 256; kb += 4) {
        int k0 = kb + 2 * half;
        v2f a  = ldv2(sh_reads + l * 256 + k0);
        v2f bf = ldv2(woutT + (32 + (k0 >> 1)) * 128 + 2 * ncol);
        acc = wmma4(a, bf, acc);
      }
      float bias = b_out[ncol];
#pragma unroll
      for (int v = 0; v < 8; ++v)
        yb[(v + 8 * half) * 64 + ncol] = acc[v] + bias;
    }
    __syncthreads();
    if (wave == 0) {
      v8f acc = {0.f, 0.f, 0.f, 0.f, 0.f, 0.f, 0.f, 0.f};
      for (int kb = 0; kb < 64; kb += 4) {
        int k0 = kb + 2 * half;
        v2f a  = ldv2(yb + l * 64 + k0);
        v2f bf = ldv2(wfcT + (k0 >> 1) * 32 + 2 * l);
        acc = wmma4(a, bf, acc);
      }
      float bias = b_fc[l];
#pragma unroll
      for (int v = 0; v < 8; ++v) {
        int row = v + 8 * half;
        out[(size_t)(b0 + row) * (NV * NTIME) + (size_t)l * NTIME + t] = acc[v] + bias;
      }
    }
    __syncthreads();
  }
}

extern "C" void kernel_launch(void* const* d_in, const int* in_sizes, int n_in,
                              void* d_out, int out_size, void* d_ws, size_t ws_size,
                              hipStream_t stream) {
  (void)in_sizes; (void)n_in; (void)out_size; (void)ws_size;
  const int*   tokens = (const int*)  d_in[0];
  const float* emb    = (const float*)d_in[1];
  const float* w_ih   = (const float*)d_in[2];
  const float* w_hh   = (const float*)d_in[3];
  const float* b_lstm = (const float*)d_in[4];
  const float* w_if   = (const float*)d_in[5];
  const float* b_if   = (const float*)d_in[6];
  const float* w_out  = (const float*)d_in[7];
  const float* b_out  = (const float*)d_in[8];
  const float* w_fc   = (const float*)d_in[9];
  const float* b_fc   = (const float*)d_in[10];

  float* ws    = (float*)d_ws;
  float* wihT  = ws;              // packed [320/2][2*256] = 81920
  float* whhT  = wihT + 81920;    // packed [64/2][2*256]  = 16384
  float* wifT  = whhT + 16384;    // packed [64/2][2*800]  = 51200
  float* woutT = wifT + 51200;    // packed [320/2][2*64]  = 20480
  float* wfcT  = woutT + 20480;   // packed [64/2][2*16]   = 1024
  float* blk   = wfcT + 1024;     // 2 blocks x (M 32768 | link 16384 | iface 12800)

  auto tr = [&](const float* s, float* d, int R, int C, int Rp) {
    int n = C * Rp;
    dnc_tr_pack2<<<(n + 255) / 256, 256, 0, stream>>>(s, d, R, C, Rp);
  };
  tr(w_ih, wihT, 256, 320, 256);
  tr(w_hh, whhT, 256, 64, 256);
  tr(w_if, wifT, 799, 64, 800);
  tr(w_out, woutT, 64, 320, 64);
  tr(w_fc, wfcT, 16, 64, 16);

  dnc_main<<<2, 256, 0, stream>>>(tokens, emb, b_lstm, b_if, b_out, b_fc,
                                  wihT, whhT, wifT, woutT, wfcT,
                                  blk, (float*)d_out);
}